// LoraSubnet_59330678227177
// MI455X (gfx1250) — compile-verified
//
#include <hip/hip_runtime.h>
#include <stdint.h>

// Problem constants (from reference): two 1024x16384 f32 tensors, keep top 10% by |x|.
#define N_ELEMS      16777216u          // 1024*16384
#define KEEP_CNT     1677722u           // N - int(0.9*N)
#define NBINS        4096
#define TIE_CAP      8192
#define HIST_BLOCKS  2048
#define HIST_ITERS   8                  // 2048 blocks * 8 iters * 256 thr * 4 elems = N
#define HIST_THREADS 256

typedef float f32x4 __attribute__((ext_vector_type(4)));

struct SelState {
  unsigned prefix;     // known high bits of threshold key (accumulates per pass)
  unsigned remaining;  // elements still to keep inside candidate set; after last pass: tie budget
  unsigned threshold;  // final 31-bit key of threshold element
  unsigned tie_count;  // number of elements whose key == threshold
  unsigned tie_cutoff; // smallest flat index among kept ties
};

// ---------------------------------------------------------------------------
// Init: zero histograms, reset per-tensor selection state (required each call:
// ws is poisoned once and never re-poisoned between graph replays).
// ---------------------------------------------------------------------------
__global__ void rsel_init(uint32_t* __restrict__ hist, SelState* __restrict__ st) {
  const int t = blockIdx.x;
  uint32_t* h = hist + t * NBINS;
  for (int i = threadIdx.x; i < NBINS; i += blockDim.x) h[i] = 0u;
  if (threadIdx.x == 0) {
    st[t].prefix = 0u;
    st[t].remaining = KEEP_CNT;
    st[t].threshold = 0u;
    st[t].tie_count = 0u;
    st[t].tie_cutoff = 0u;
  }
}

// ---------------------------------------------------------------------------
// Histogram pass: stream input through LDS via gfx1250 async global->LDS DMA
// (double-buffered, ASYNCcnt-tracked), bin candidates into an LDS histogram
// (ds_add_u32 atomics), then merge to the global histogram.
// key = f32 bits & 0x7FFFFFFF  (monotone in |x| for non-negative floats).
// Candidate iff (key >> (shift+bits)) == state.prefix.
// ---------------------------------------------------------------------------
__global__ __launch_bounds__(HIST_THREADS)
void rsel_hist(const uint32_t* __restrict__ A, const uint32_t* __restrict__ B,
               uint32_t* __restrict__ hist, const SelState* __restrict__ st,
               int shift, int bits)
{
  const int t = blockIdx.y;
  const uint32_t* __restrict__ src = t ? B : A;
  uint32_t* __restrict__ gh = hist + t * NBINS;
  const unsigned prefix  = st[t].prefix;
  const unsigned binmask = (1u << bits) - 1u;
  const int      hishift = shift + bits;   // <= 31 always

  __shared__ uint32_t sh[NBINS];                      // 16 KB histogram
  __shared__ uint32_t stage[2][HIST_THREADS * 4];     // 8 KB double-buffer staging

  for (int i = threadIdx.x; i < NBINS; i += HIST_THREADS) sh[i] = 0u;
  __syncthreads();

  // Generic LDS pointers carry the LDS byte address in their low 32 bits
  // (ISA 10.2) -> usable directly as async VDST.
  const unsigned long long gbase = (unsigned long long)(uintptr_t)src;
  const unsigned lds0 = (unsigned)(uintptr_t)&stage[0][threadIdx.x * 4];
  const unsigned lds1 = (unsigned)(uintptr_t)&stage[1][threadIdx.x * 4];
  const unsigned gofs0 =
      ((unsigned)blockIdx.x * (HIST_ITERS * HIST_THREADS * 4u) +
       (unsigned)threadIdx.x * 4u) * 4u;              // byte offset (GVS: saddr + voffset)

  // Prologue: kick off buffer 0.
  asm volatile("global_load_async_to_lds_b128 %0, %1, %2"
               :: "v"(lds0), "v"(gofs0), "s"(gbase) : "memory");

  for (int i = 0; i < HIST_ITERS; ++i) {
    if (i + 1 < HIST_ITERS) {
      const unsigned nofs = gofs0 + (unsigned)(i + 1) * (HIST_THREADS * 16u);
      const unsigned nlds = ((i + 1) & 1) ? lds1 : lds0;
      asm volatile("global_load_async_to_lds_b128 %0, %1, %2"
                   :: "v"(nlds), "v"(nofs), "s"(gbase) : "memory");
      // two outstanding: wait until the older (current buffer) has landed
      asm volatile("s_wait_asynccnt 0x1" ::: "memory");
    } else {
      asm volatile("s_wait_asynccnt 0x0" ::: "memory");
    }
    const uint32_t* sp = (i & 1) ? &stage[1][threadIdx.x * 4]
                                 : &stage[0][threadIdx.x * 4];
#pragma unroll
    for (int e = 0; e < 4; ++e) {
      const unsigned key = sp[e] & 0x7FFFFFFFu;
      if ((key >> hishift) == prefix)
        atomicAdd(&sh[(key >> shift) & binmask], 1u);   // ds_add_u32
    }
  }

  __syncthreads();
  for (int i = threadIdx.x; i < NBINS; i += HIST_THREADS) {
    const unsigned c = sh[i];
    if (c) atomicAdd(&gh[i], c);
  }
}

// ---------------------------------------------------------------------------
// Scan pass: one workgroup per tensor. Suffix-sum the 4096-bin histogram from
// the top, locate the bin containing the remaining-th largest key, update the
// prefix / remaining count, and zero the histogram for the next pass.
// ---------------------------------------------------------------------------
__global__ __launch_bounds__(1024)
void rsel_scan(uint32_t* __restrict__ hist, SelState* __restrict__ st,
               int bits, int is_last)
{
  const int t = blockIdx.y;
  uint32_t* __restrict__ gh = hist + t * NBINS;
  SelState* __restrict__ s = st + t;
  const int tid = threadIdx.x;                 // 1024 threads, 4 bins each

  const uint32_t v0 = gh[4*tid+0], v1 = gh[4*tid+1],
                 v2 = gh[4*tid+2], v3 = gh[4*tid+3];
  const uint32_t chunk = v0 + v1 + v2 + v3;

  __shared__ uint32_t csum[1024];
  csum[tid] = chunk;
  const unsigned rem = s->remaining;           // snapshot before anyone writes
  const unsigned pfx = s->prefix;
  __syncthreads();

  // Hillis-Steele inclusive suffix scan over chunk sums.
  for (int off = 1; off < 1024; off <<= 1) {
    const uint32_t add = (tid + off < 1024) ? csum[tid + off] : 0u;
    __syncthreads();
    csum[tid] += add;
    __syncthreads();
  }

  unsigned a = csum[tid] - chunk;              // count in bins strictly above my chunk
  const uint32_t hv[4] = {v0, v1, v2, v3};
#pragma unroll
  for (int e = 3; e >= 0; --e) {               // descending bins within my chunk
    const unsigned h = hv[e];
    if (a < rem && a + h >= rem) {             // exactly one thread/bin hits this
      const unsigned np = (pfx << bits) | (unsigned)(4*tid + e);
      s->prefix = np;
      s->remaining = rem - a;                  // elements to take inside this bin
      if (is_last) s->threshold = np;          // full 31-bit key after 12+12+7 bits
    }
    a += h;
  }

  // Reset histogram for the next pass.
  gh[4*tid+0] = 0u; gh[4*tid+1] = 0u; gh[4*tid+2] = 0u; gh[4*tid+3] = 0u;
}

// ---------------------------------------------------------------------------
// Tie gather: collect flat indices of elements whose key equals the threshold
// (input is L2-resident by now). Count is tiny for random data.
// ---------------------------------------------------------------------------
__global__ __launch_bounds__(256)
void rsel_tie_gather(const uint32_t* __restrict__ A, const uint32_t* __restrict__ B,
                     SelState* __restrict__ st, uint32_t* __restrict__ tiebuf)
{
  const int t = blockIdx.y;
  const uint32_t* __restrict__ src = t ? B : A;
  SelState* __restrict__ s = st + t;
  const unsigned T = s->threshold;
  uint32_t* __restrict__ tb = tiebuf + t * TIE_CAP;
  const unsigned nthreads = gridDim.x * blockDim.x;

  for (unsigned q = blockIdx.x * blockDim.x + threadIdx.x; q < N_ELEMS / 4u; q += nthreads) {
    const uint4 v = ((const uint4*)src)[q];
    const unsigned base = q * 4u;
    const unsigned k0 = v.x & 0x7FFFFFFFu, k1 = v.y & 0x7FFFFFFFu,
                   k2 = v.z & 0x7FFFFFFFu, k3 = v.w & 0x7FFFFFFFu;
    if (k0 == T) { unsigned p = atomicAdd(&s->tie_count, 1u); if (p < TIE_CAP) tb[p] = base + 0u; }
    if (k1 == T) { unsigned p = atomicAdd(&s->tie_count, 1u); if (p < TIE_CAP) tb[p] = base + 1u; }
    if (k2 == T) { unsigned p = atomicAdd(&s->tie_count, 1u); if (p < TIE_CAP) tb[p] = base + 2u; }
    if (k3 == T) { unsigned p = atomicAdd(&s->tie_count, 1u); if (p < TIE_CAP) tb[p] = base + 3u; }
  }
}

// ---------------------------------------------------------------------------
// Tie cutoff: stable argsort keeps the ties with the LARGEST flat indices.
// cutoff = tie_keep-th largest tie index (rank computed set-wise, so the
// result is independent of the nondeterministic gather order).
// ---------------------------------------------------------------------------
__global__ __launch_bounds__(256)
void rsel_tie_cutoff(SelState* __restrict__ st, const uint32_t* __restrict__ tiebuf)
{
  const int t = blockIdx.y;
  SelState* __restrict__ s = st + t;
  unsigned tc = s->tie_count; if (tc > TIE_CAP) tc = TIE_CAP;
  const unsigned keep = s->remaining;          // tie budget, >= 1 by construction
  const uint32_t* __restrict__ tb = tiebuf + t * TIE_CAP;

  if (keep >= tc) { if (threadIdx.x == 0) s->tie_cutoff = 0u; return; }

  for (unsigned i = threadIdx.x; i < tc; i += blockDim.x) {
    const unsigned x = tb[i];
    unsigned r = 0;
    for (unsigned j = 0; j < tc; ++j) r += (tb[j] > x) ? 1u : 0u;
    if (r == keep - 1u) s->tie_cutoff = x;     // unique: indices are distinct
  }
}

// ---------------------------------------------------------------------------
// Mask write: read L2-resident input, emit 1.0f/0.0f. Non-temporal b128
// stores keep the write-once output from evicting the inputs from L2;
// global_prefetch keeps the read stream ahead.
// ---------------------------------------------------------------------------
__global__ __launch_bounds__(256)
void rsel_mask(const uint32_t* __restrict__ A, const uint32_t* __restrict__ B,
               const SelState* __restrict__ st, float* __restrict__ out)
{
  const int t = blockIdx.y;
  const uint32_t* __restrict__ src = t ? B : A;
  float* __restrict__ dst = out + (size_t)t * N_ELEMS;
  const unsigned T   = st[t].threshold;
  const unsigned cut = st[t].tie_cutoff;
  const unsigned nthreads = gridDim.x * blockDim.x;

  for (unsigned q = blockIdx.x * blockDim.x + threadIdx.x; q < N_ELEMS / 4u; q += nthreads) {
    __builtin_prefetch(src + (size_t)q * 4u + (size_t)nthreads * 4u, 0, 0);
    const uint4 v = ((const uint4*)src)[q];
    const unsigned base = q * 4u;
    f32x4 m;
    unsigned k;
    k = v.x & 0x7FFFFFFFu; m.x = (k > T || (k == T && base + 0u >= cut)) ? 1.0f : 0.0f;
    k = v.y & 0x7FFFFFFFu; m.y = (k > T || (k == T && base + 1u >= cut)) ? 1.0f : 0.0f;
    k = v.z & 0x7FFFFFFFu; m.z = (k > T || (k == T && base + 2u >= cut)) ? 1.0f : 0.0f;
    k = v.w & 0x7FFFFFFFu; m.w = (k > T || (k == T && base + 3u >= cut)) ? 1.0f : 0.0f;
    __builtin_nontemporal_store(m, (f32x4*)dst + q);
  }
}

// ---------------------------------------------------------------------------
extern "C" void kernel_launch(void* const* d_in, const int* in_sizes, int n_in,
                              void* d_out, int out_size, void* d_ws, size_t ws_size,
                              hipStream_t stream)
{
  (void)in_sizes; (void)n_in; (void)out_size; (void)ws_size;

  const uint32_t* A = (const uint32_t*)d_in[0];   // lora_A_mask (f32 bits)
  const uint32_t* B = (const uint32_t*)d_in[1];   // lora_B_mask (f32 bits)
  float* out = (float*)d_out;                     // [mask_A | mask_B]

  char* ws = (char*)d_ws;
  uint32_t* hist   = (uint32_t*)ws;               // 2 * 4096 * 4B = 32 KB
  SelState* st     = (SelState*)(ws + 32768);     // 2 * 20 B
  uint32_t* tiebuf = (uint32_t*)(ws + 32768 + 256); // 2 * 8192 * 4B = 64 KB

  rsel_init<<<dim3(2), 1024, 0, stream>>>(hist, st);

  // 31-bit radix select over |x| bit patterns: 12 + 12 + 7 bits.
  rsel_hist<<<dim3(HIST_BLOCKS, 2), HIST_THREADS, 0, stream>>>(A, B, hist, st, 19, 12);
  rsel_scan<<<dim3(1, 2), 1024, 0, stream>>>(hist, st, 12, 0);
  rsel_hist<<<dim3(HIST_BLOCKS, 2), HIST_THREADS, 0, stream>>>(A, B, hist, st, 7, 12);
  rsel_scan<<<dim3(1, 2), 1024, 0, stream>>>(hist, st, 12, 0);
  rsel_hist<<<dim3(HIST_BLOCKS, 2), HIST_THREADS, 0, stream>>>(A, B, hist, st, 0, 7);
  rsel_scan<<<dim3(1, 2), 1024, 0, stream>>>(hist, st, 7, 1);

  rsel_tie_gather<<<dim3(1024, 2), 256, 0, stream>>>(A, B, st, tiebuf);
  rsel_tie_cutoff<<<dim3(1, 2), 256, 0, stream>>>(st, tiebuf);

  rsel_mask<<<dim3(1024, 2), 256, 0, stream>>>(A, B, st, out);
}